// SceneSegmentationModel_55997783605450
// MI455X (gfx1250) — compile-verified
//
#include <hip/hip_runtime.h>
#include <math.h>

typedef __attribute__((ext_vector_type(2))) float v2f;
typedef __attribute__((ext_vector_type(8))) float v8f;

#define B_ 4
#define C_ 144
#define N_ 16384
#define K_ 16
#define G_ 9

#define WV_PER_BLK 8
#define COL_STRIDE 20          // dwords per staged column (16 data + 4 pad; 16B align, no bank conflicts)
#define LDS_PER_WAVE 672       // 320 key + 320 val + 16 feat + 16 pad (dwords)

// ---------------------------------------------------------------------------
// Transpose (BG, 16, N) -> (BG, N, 16): makes every neighbor gather a single
// contiguous 64B access (4 x b128) instead of 16 loads strided by 64KB.
// ---------------------------------------------------------------------------
__global__ void __launch_bounds__(256) gtl_transpose_kernel(const float* __restrict__ in,
                                                            float* __restrict__ out) {
    __shared__ float t[16 * 256];
    const int bg  = blockIdx.y;
    const int n0  = blockIdx.x * 256;
    const int tid = threadIdx.x;

    const float* src = in + (size_t)bg * 16 * N_ + n0;
#pragma unroll
    for (int d = 0; d < 16; ++d)                 // coalesced reads across tid
        t[d * 256 + tid] = src[(size_t)d * N_ + tid];
    __syncthreads();

    float* dst = out + ((size_t)bg * N_ + n0 + tid) * 16;
#pragma unroll
    for (int i = 0; i < 4; ++i) {                // 64B contiguous per thread
        float4 o;
        o.x = t[(4 * i + 0) * 256 + tid];
        o.y = t[(4 * i + 1) * 256 + tid];
        o.z = t[(4 * i + 2) * 256 + tid];
        o.w = t[(4 * i + 3) * 256 + tid];
        ((float4*)dst)[i] = o;
    }
}

__global__ void __launch_bounds__(256) gtl_zero_kernel(float* __restrict__ p, int n) {
    int i = blockIdx.x * blockDim.x + threadIdx.x;
    if (i < n) p[i] = 0.0f;
}

// ---------------------------------------------------------------------------
// Main kernel: one wave32 per (b,g,n).
// Lanes 0-15 gather the 16 key columns, lanes 16-31 the 16 value columns.
// att  = 4 x V_WMMA_F32_16X16X4_F32 (A = Q replicated rows, B = key tile)
// feat = 4 x V_WMMA_F32_16X16X4_F32 (A = value tile, B = weights broadcast)
// ---------------------------------------------------------------------------
__global__ void __launch_bounds__(256) gtl_main_kernel(const float* __restrict__ qT,
                                                       const float* __restrict__ vT,
                                                       const int* __restrict__ idx,
                                                       float* __restrict__ feat,
                                                       float* __restrict__ cent) {
    __shared__ float lds[WV_PER_BLK * LDS_PER_WAVE];

    const int tid  = threadIdx.x;
    const int wv   = tid >> 5;
    const int lane = tid & 31;
    const int half = lane >> 4;   // 0: lanes 0-15, 1: lanes 16-31
    const int j    = lane & 15;

    const int nid = blockIdx.x * WV_PER_BLK + wv;   // flat (b,g,n)
    const int n   = nid % N_;
    const int bg  = nid / N_;                       // b*G + g

    const float* qTb  = qT + (size_t)bg * N_ * 16;
    const float* vTb  = vT + (size_t)bg * N_ * 16;
    const int*   idxn = idx + ((size_t)(bg / G_) * N_ + n) * K_;

    float* ldsW    = lds + wv * LDS_PER_WAVE;
    float* ldsKey  = ldsW;
    float* ldsVal  = ldsW + 16 * COL_STRIDE;
    float* ldsFeat = ldsW + 32 * COL_STRIDE;

    // ---- gather one column per lane (64B contiguous, thanks to transpose) ----
    const int col = idxn[j];
    const float4* gsrc = (const float4*)((half ? vTb : qTb) + (size_t)col * 16);
    float4 c0 = gsrc[0], c1 = gsrc[1], c2 = gsrc[2], c3 = gsrc[3];

    float4* stc = (float4*)((half ? ldsVal : ldsKey) + j * COL_STRIDE);
    stc[0] = c0; stc[1] = c1; stc[2] = c2; stc[3] = c3;

    // ---- A operand for att: load exactly the two q elements this lane needs
    // per K-slice as a float2 (q[4c+2h], q[4c+2h+1]) -> global_load_b64,
    // same 64B line for the whole wave. No register array, no dynamic extracts.
    const v2f* qp2 = (const v2f*)(qTb + (size_t)n * 16);
    v2f qa[4];
#pragma unroll
    for (int c = 0; c < 4; ++c) qa[c] = qp2[2 * c + half];

    // staged tiles must be visible to all lanes of this wave (LDS is in-order)
    asm volatile("s_wait_dscnt 0" ::: "memory");

    // ---- att[j] = sum_d q[d] * key[d][j] via 4 chained f32 WMMAs ----
    v8f acc = {};
#pragma unroll
    for (int c = 0; c < 4; ++c) {
        // B[K][j]: key tile column j, elements 4c+2h, 4c+2h+1 (8B aligned)
        v2f b = *(const v2f*)&ldsKey[j * COL_STRIDE + 4 * c + 2 * half];
        acc = __builtin_amdgcn_wmma_f32_16x16x4_f32(false, qa[c], false, b,
                                                    (short)0, acc, false, false);
    }
    float att = acc[0];                         // att[lane % 16], rows identical

    // ---- softmax over the 16 neighbors (within each 16-lane group) ----
    float mx = att;
#pragma unroll
    for (int m = 1; m < 16; m <<= 1) mx = fmaxf(mx, __shfl_xor(mx, m, 32));
    float e = __expf(att - mx);
    float s = e;
#pragma unroll
    for (int m = 1; m < 16; m <<= 1) s += __shfl_xor(s, m, 32);
    float w = e / s;

    // ---- attention centrality: scatter-add weights onto neighbor ids ----
    if (!half) atomicAdd(cent + (size_t)bg * N_ + col, w);

    // ---- feat[d] = sum_k val[d][k] * w[k] via 4 chained f32 WMMAs ----
    v8f fa = {};
#pragma unroll
    for (int c = 0; c < 4; ++c) {
        v2f a;                                  // A[m=d][K=k]: value tile
        a.x = ldsVal[(4 * c + 2 * half + 0) * COL_STRIDE + j];
        a.y = ldsVal[(4 * c + 2 * half + 1) * COL_STRIDE + j];
        v2f b;                                  // B[K][j]: w[k] broadcast per row
        b.x = __shfl(w, 4 * c + 2 * half + 0, 32);
        b.y = __shfl(w, 4 * c + 2 * half + 1, 32);
        fa = __builtin_amdgcn_wmma_f32_16x16x4_f32(false, a, false, b,
                                                   (short)0, fa, false, false);
    }

    // ---- result is replicated across columns: stage rows 0-7 / 8-15 ----
    if (j == 0) {
        float4 f0; f0.x = fa[0]; f0.y = fa[1]; f0.z = fa[2]; f0.w = fa[3];
        float4 f1; f1.x = fa[4]; f1.y = fa[5]; f1.z = fa[6]; f1.w = fa[7];
        float4* fp = (float4*)(ldsFeat + 8 * half);
        fp[0] = f0; fp[1] = f1;
    }
    asm volatile("s_wait_dscnt 0" ::: "memory");

    if (!half) {
        // out index: (b*C + g*16 + d)*N + n == (bg*16 + d)*N + n  (C = G*16)
        feat[((size_t)bg * 16 + j) * N_ + n] = ldsFeat[j];
    }
}

// ---------------------------------------------------------------------------
extern "C" void kernel_launch(void* const* d_in, const int* in_sizes, int n_in,
                              void* d_out, int out_size, void* d_ws, size_t ws_size,
                              hipStream_t stream) {
    const float* qk  = (const float*)d_in[0];   // (B, C, N) f32
    const float* val = (const float*)d_in[1];   // (B, C, N) f32
    const int*   idx = (const int*)d_in[2];     // (B, N, K) i32

    float* out   = (float*)d_out;
    float* featp = out;                                   // (B, C, N)
    float* centp = out + (size_t)B_ * C_ * N_;            // (B, G, N)

    float* qT = (float*)d_ws;                             // (BG, N, 16)
    float* vT = qT + (size_t)B_ * G_ * N_ * 16;           // (BG, N, 16)

    dim3 tb(256);
    dim3 tg(N_ / 256, B_ * G_);
    gtl_transpose_kernel<<<tg, tb, 0, stream>>>(qk, qT);
    gtl_transpose_kernel<<<tg, tb, 0, stream>>>(val, vT);

    const int centN = B_ * G_ * N_;
    gtl_zero_kernel<<<(centN + 255) / 256, 256, 0, stream>>>(centp, centN);

    const int nPoints = B_ * G_ * N_;
    gtl_main_kernel<<<nPoints / WV_PER_BLK, 256, 0, stream>>>(qT, vT, idx, featp, centp);
}